// GSNConv_11622181503635
// MI455X (gfx1250) — compile-verified
//
#include <hip/hip_runtime.h>
#include <hip/hip_bf16.h>

typedef float v2f __attribute__((ext_vector_type(2)));
typedef float v8f __attribute__((ext_vector_type(8)));

#define D_FEAT 64
#define D_HID  128
#define WAVES  8

// Guaranteed CDNA5 hardware float atomic: GLOBAL_ATOMIC_ADD_F32, no return
// (tracked by STOREcnt; S_ENDPGM's implicit wait-idle covers completion).
__device__ __forceinline__ void atomic_add_f32_hw(float* p, float v) {
  asm volatile("global_atomic_add_f32 %0, %1, off"
               :
               : "v"(p), "v"(v)
               : "memory");
}

// ---------------------------------------------------------------------------
// Kernel 1: h = (1 + eps) * x   (float4 vectorized, re-inits ws every call)
// ---------------------------------------------------------------------------
__global__ void init_h_kernel(const float* __restrict__ x,
                              const float* __restrict__ eps,
                              float* __restrict__ h, int n4) {
  int i = blockIdx.x * blockDim.x + threadIdx.x;
  if (i < n4) {
    float s = 1.0f + eps[0];
    float4 v = ((const float4*)x)[i];
    v.x *= s; v.y *= s; v.z *= s; v.w *= s;
    ((float4*)h)[i] = v;
  }
}

// ---------------------------------------------------------------------------
// Kernel 2: scatter-add. 16 threads per edge, each owns a float4 chunk.
// Gather is 256B-contiguous per edge; atomics are hw global_atomic_add_f32
// resolving in L2 (25.6MB agg buffer << 192MB L2).
// ---------------------------------------------------------------------------
__global__ void scatter_edges_kernel(const float* __restrict__ x,
                                     const int* __restrict__ esrc,
                                     const int* __restrict__ edst,
                                     float* __restrict__ h, int nEdges) {
  int tid = blockIdx.x * blockDim.x + threadIdx.x;
  int e = tid >> 4;
  if (e >= nEdges) return;
  int c = (tid & 15) * 4;
  int s = esrc[e];
  int d = edst[e];
  float4 v = *(const float4*)(x + (size_t)s * D_FEAT + c);
  float* p = h + (size_t)d * D_FEAT + c;
  atomic_add_f32_hw(p + 0, v.x);
  atomic_add_f32_hw(p + 1, v.y);
  atomic_add_f32_hw(p + 2, v.z);
  atomic_add_f32_hw(p + 3, v.w);
}

// ---------------------------------------------------------------------------
// Kernel 3: fused MLP  out = relu(h@W1 + b1)@W2 + b2  using fp32 WMMA.
// One wave computes a 16-row tile. 8 waves / 256-thread block.
// GEMM1 A-frags from global (rows stay in WGP$ across k-steps); h1 staged
// through a per-wave XOR-swizzled LDS tile for the GEMM2 cross-lane layout.
// ---------------------------------------------------------------------------
__global__ __launch_bounds__(256) void gin_mlp_kernel(
    const float* __restrict__ h,
    const float* __restrict__ W1, const float* __restrict__ b1,
    const float* __restrict__ W2, const float* __restrict__ b2,
    float* __restrict__ out, int nNodes) {
  __shared__ float sH1[WAVES][16 * D_HID];   // 64KB total

  const int lane = threadIdx.x & 31;
  const int wave = threadIdx.x >> 5;
  const int ln = lane & 15;   // N (or M-row for A frags)
  const int hb = lane >> 4;   // half-wave: selects K pair / M+8 half
  const int rowBase = (blockIdx.x * WAVES + wave) * 16;

  // A-matrix row this lane feeds (clamped; stores are guarded later)
  int arow = rowBase + ln;
  if (arow >= nNodes) arow = nNodes - 1;
  const float* aptr = h + (size_t)arow * D_FEAT + hb * 2;

  const v8f vzero = {0.f, 0.f, 0.f, 0.f, 0.f, 0.f, 0.f, 0.f};

  // ---- GEMM1: [16 x 64] @ [64 x 128] -> 8 accumulator tiles -------------
  v8f acc1[8];
#pragma unroll
  for (int t = 0; t < 8; ++t) acc1[t] = vzero;

  for (int s = 0; s < 16; ++s) {          // K = 64, 4 per wmma
    // A frag: lanes 0-15 hold K=4s,4s+1 ; lanes 16-31 hold K=4s+2,4s+3
    v2f a = *(const v2f*)(aptr + s * 4);
    const int k0 = 4 * s + 2 * hb;
#pragma unroll
    for (int t = 0; t < 8; ++t) {
      v2f b;
      b.x = W1[(k0 + 0) * D_HID + t * 16 + ln];
      b.y = W1[(k0 + 1) * D_HID + t * 16 + ln];
      acc1[t] = __builtin_amdgcn_wmma_f32_16x16x4_f32(
          false, a, false, b, (short)0, acc1[t], false, false);
    }
  }

  // ---- bias + ReLU, stage h1 into swizzled LDS --------------------------
  float* myH1 = &sH1[wave][0];
#pragma unroll
  for (int t = 0; t < 8; ++t) {
#pragma unroll
    for (int j = 0; j < 8; ++j) {
      int m = j + hb * 8;                 // C layout: VGPR j -> M=j / j+8
      int n = t * 16 + ln;
      float v = acc1[t][j] + b1[n];
      v = v > 0.f ? v : 0.f;
      myH1[m * D_HID + (n ^ (4 * m))] = v;   // XOR swizzle: conflict-free
    }
  }
  __syncthreads();

  // ---- GEMM2: [16 x 128] @ [128 x 64] -> 4 accumulator tiles ------------
  v8f acc2[4];
#pragma unroll
  for (int t = 0; t < 4; ++t) acc2[t] = vzero;

  for (int s = 0; s < 32; ++s) {          // K = 128, 4 per wmma
    const int c = 4 * s + 2 * hb;
    // swizzle keeps the float2 contiguous (xor value has no low bits set)
    v2f a = *(const v2f*)&myH1[ln * D_HID + (c ^ (4 * ln))];
#pragma unroll
    for (int t = 0; t < 4; ++t) {
      v2f b;
      b.x = W2[(c + 0) * D_FEAT + t * 16 + ln];
      b.y = W2[(c + 1) * D_FEAT + t * 16 + ln];
      acc2[t] = __builtin_amdgcn_wmma_f32_16x16x4_f32(
          false, a, false, b, (short)0, acc2[t], false, false);
    }
  }

  // ---- bias + guarded store --------------------------------------------
#pragma unroll
  for (int t = 0; t < 4; ++t) {
#pragma unroll
    for (int j = 0; j < 8; ++j) {
      int m = j + hb * 8;
      int g = rowBase + m;
      int n = t * 16 + ln;
      if (g < nNodes) out[(size_t)g * D_FEAT + n] = acc2[t][j] + b2[n];
    }
  }
}

// ---------------------------------------------------------------------------
extern "C" void kernel_launch(void* const* d_in, const int* in_sizes, int n_in,
                              void* d_out, int out_size, void* d_ws, size_t ws_size,
                              hipStream_t stream) {
  const float* node_attr = (const float*)d_in[0];
  const float* W1  = (const float*)d_in[1];
  const float* b1  = (const float*)d_in[2];
  const float* W2  = (const float*)d_in[3];
  const float* b2  = (const float*)d_in[4];
  const float* eps = (const float*)d_in[5];
  const int* esrc  = (const int*)d_in[6];
  const int* edst  = (const int*)d_in[7];
  float* out = (float*)d_out;

  const int nNodes = in_sizes[0] / D_FEAT;
  const int nEdges = in_sizes[6];

  const size_t need = (size_t)nNodes * D_FEAT * sizeof(float);
  float* hbuf = (ws_size >= need) ? (float*)d_ws : out;  // in-place fallback is safe

  // 1) h = (1+eps)*x
  const int n4 = nNodes * (D_FEAT / 4);
  init_h_kernel<<<(n4 + 255) / 256, 256, 0, stream>>>(node_attr, eps, hbuf, n4);

  // 2) scatter-add over edges
  const long long sThreads = (long long)nEdges * 16;
  scatter_edges_kernel<<<(int)((sThreads + 255) / 256), 256, 0, stream>>>(
      node_attr, esrc, edst, hbuf, nEdges);

  // 3) fused WMMA MLP
  const int tiles = (nNodes + 15) / 16;
  const int blocks = (tiles + WAVES - 1) / WAVES;
  gin_mlp_kernel<<<blocks, 256, 0, stream>>>(hbuf, W1, b1, W2, b2, out, nNodes);
}